// TangentProjections_89369679495272
// MI455X (gfx1250) — compile-verified
//
#include <hip/hip_runtime.h>
#include <math.h>

typedef __attribute__((ext_vector_type(2))) float v2f;
typedef __attribute__((ext_vector_type(8))) float v8f;
typedef __attribute__((ext_vector_type(4))) unsigned int v4u;
typedef __attribute__((ext_vector_type(8))) int v8i;
typedef __attribute__((ext_vector_type(4))) int v4i;

#define BATCH 4
#define NPTS  4096
#define KNB   32          // neighbors kept
#define LIST  (KNB + 1)   // +1 for radius entry
#define NTILE (NPTS / 16) // 256 column tiles
#define NFLT  (NPTS * 3)  // floats per batch = 12288

__global__ __launch_bounds__(32)
void TangentProjections_89369679495272_kernel(const float* __restrict__ coords,
                                              float* __restrict__ out)
{
    // One wave (32 lanes) handles 16 rows of one batch.
    const int wg   = blockIdx.x;
    const int b    = wg / (NPTS / 16);
    const int rt   = wg % (NPTS / 16);
    const int row_base = rt * 16;

    const int lane = threadIdx.x;
    const int ml   = lane & 15;       // row (A) / col (B) within tile
    const bool hi  = lane >= 16;      // upper half-wave

    const float* base = coords + (size_t)b * NFLT;

    __shared__ float  s_pts[NFLT];            // whole batch staged by TDM (48 KB)
    __shared__ float  s_rowsq[16];
    __shared__ float  s_dtile[16][16];
    __shared__ float  s_list_d[16][LIST];
    __shared__ int    s_list_i[16][LIST];
    __shared__ float4 s_nb[16][KNB];

    // ================= TDM: stage the whole batch's coords into LDS =========
    // D# per CDNA5 ISA ch.8: 1D tile of 12288 x 4-byte elements (groups 2/3
    // unused: tile_dim2/3/4 = 0).  6-arg builtin form (clang-23 flavor).
    {
        const unsigned long long gaddr = (unsigned long long)(size_t)base;
        const unsigned int lds_off = (unsigned int)(size_t)(&s_pts[0]);

        v4u g0;
        g0[0] = 1u;                                        // count=1, user desc
        g0[1] = lds_off;                                   // lds_addr
        g0[2] = (unsigned int)(gaddr & 0xffffffffu);       // global_addr[31:0]
        g0[3] = (unsigned int)((gaddr >> 32) & 0x01ffffffu)
              | 0x80000000u;                               // addr[56:32] | type=2

        v8i g1;
        g1[0] = (int)0x00020000;      // wg_mask=0, data_size=2 (4B), no flags
        g1[1] = (int)0x30000000;      // tensor_dim0[15:0]=12288 in bits[31:16]
        g1[2] = (int)0x00010000;      // tensor_dim0[31:16]=0, tensor_dim1[15:0]=1
        g1[3] = (int)0x30000000;      // tensor_dim1[31:16]=0, tile_dim0=12288
        g1[4] = (int)0x00000001;      // tile_dim1=1, tile_dim2=0
        g1[5] = (int)12288;           // tensor_dim0_stride[31:0]
        g1[6] = (int)0x30000000;      // stride0[47:32]=0, tensor_dim1_stride[15:0]=12288
        g1[7] = 0;                    // tensor_dim1_stride[47:16]=0

        v4i g2 = {0, 0, 0, 0};        // dims 2/3 unused (tile_dim3=0)
        v4i g3 = {0, 0, 0, 0};        // dim 4 unused
        v8i g4 = {0, 0, 0, 0, 0, 0, 0, 0};  // unused trailing group (zero-filled)

        __builtin_amdgcn_tensor_load_to_lds(g0, g1, g2, g3, g4, /*cpol=*/0);
    }
    __builtin_amdgcn_s_wait_tensorcnt(0);   // data resident in LDS for this wave

    // ---- A fragment: 16x4 f32 (rows = this tile's points, K = x,y,z,0) ----
    // lanes 0-15: VGPR0=K0(x), VGPR1=K1(y);  lanes 16-31: VGPR0=K2(z), VGPR1=K3(0)
    const float* rp = &s_pts[(row_base + ml) * 3];
    const float rx = rp[0], ry = rp[1], rz = rp[2];
    v2f af;
    af[0] = hi ? rz : rx;
    af[1] = hi ? 0.0f : ry;
    if (!hi) s_rowsq[ml] = rx * rx + ry * ry + rz * rz;
    __syncthreads();

    // row squared-norms for the 8 M values this lane's accumulator holds
    float rsq[8];
#pragma unroll
    for (int v = 0; v < 8; ++v) rsq[v] = s_rowsq[(hi ? 8 : 0) + v];

    int cnt = 0;  // per-row list fill count (owned by lanes 0-15)

    for (int t = 0; t < NTILE; ++t) {
        const int col_base = t * 16;
        const float* cp = &s_pts[(col_base + ml) * 3];
        const float cx = cp[0], cy = cp[1], cz = cp[2];
        const float cs = cx * cx + cy * cy + cz * cz;

        // ---- B fragment: 4x16 f32, mirrored layout of A ----
        v2f bf;
        bf[0] = hi ? cz : cx;
        bf[1] = hi ? 0.0f : cy;

        v8f acc = {};
        acc = __builtin_amdgcn_wmma_f32_16x16x4_f32(
            /*neg_a=*/false, af, /*neg_b=*/false, bf,
            /*c_mod=*/(short)0, acc, /*reuse_a=*/false, /*reuse_b=*/false);

        // d(m,n) = sqrt(max(|m|^2 + |n|^2 - 2 m.n, 0)); scatter tile to LDS
#pragma unroll
        for (int v = 0; v < 8; ++v) {
            const float d2 = rsq[v] + cs - 2.0f * acc[v];
            const float d  = (d2 > 0.0f) ? sqrtf(d2) : 0.0f;
            s_dtile[(hi ? 8 : 0) + v][ml] = d;
        }
        __syncthreads();

        // ---- streaming top-33 insertion, one row per lane (lanes 0-15) ----
        if (!hi) {
            const int m = ml;
            for (int c = 0; c < 16; ++c) {
                const float d = s_dtile[m][c];
                const int idx = col_base + c;
                if (cnt < LIST) {
                    int p = cnt++;
                    while (p > 0 && s_list_d[m][p - 1] > d) {
                        s_list_d[m][p] = s_list_d[m][p - 1];
                        s_list_i[m][p] = s_list_i[m][p - 1];
                        --p;
                    }
                    s_list_d[m][p] = d;
                    s_list_i[m][p] = idx;
                } else if (d < s_list_d[m][LIST - 1]) {
                    int p = LIST - 1;
                    while (p > 0 && s_list_d[m][p - 1] > d) {
                        s_list_d[m][p] = s_list_d[m][p - 1];
                        s_list_i[m][p] = s_list_i[m][p - 1];
                        --p;
                    }
                    s_list_d[m][p] = d;
                    s_list_i[m][p] = idx;
                }
            }
        }
        __syncthreads();
    }

    // ================= Phase 3: per-row local frame + projection =============
    if (!hi) {
        const int m   = ml;
        const int row = row_base + m;
        const float radius = s_list_d[m][KNB];   // 33rd smallest

        const float px = rx, py = ry, pz = rz;   // this row's point

        // weighted covariance of neighbor offsets (neighbors read from LDS)
        float cxx = 0.f, cxy = 0.f, cxz = 0.f, cyy = 0.f, cyz = 0.f, czz = 0.f;
        float wsum = 0.f;
        for (int j = 0; j < KNB; ++j) {
            const int idx = s_list_i[m][j];
            const float* qp = &s_pts[idx * 3];
            const float nx = qp[0] - px, ny = qp[1] - py, nz = qp[2] - pz;
            const float dj = s_list_d[m][j];
            s_nb[m][j] = make_float4(nx, ny, nz, dj);
            const float w = radius - dj;
            wsum += w;
            cxx += w * nx * nx; cxy += w * nx * ny; cxz += w * nx * nz;
            cyy += w * ny * ny; cyz += w * ny * nz; czz += w * nz * nz;
        }
        const float inv = 1.0f / wsum;
        float a00 = cxx * inv, a01 = cxy * inv, a02 = cxz * inv;
        float a11 = cyy * inv, a12 = cyz * inv, a22 = czz * inv;

        // ---- 3x3 symmetric Jacobi eigensolver (8 sweeps) ----
        float v00 = 1.f, v01 = 0.f, v02 = 0.f;
        float v10 = 0.f, v11 = 1.f, v12 = 0.f;
        float v20 = 0.f, v21 = 0.f, v22 = 1.f;
        for (int sweep = 0; sweep < 8; ++sweep) {
            // (p,q) = (0,1), r = 2
            {
                const float apq = a01;
                if (fabsf(apq) > 1e-30f) {
                    const float theta = (a11 - a00) / (2.0f * apq);
                    float tt = 1.0f / (fabsf(theta) + sqrtf(theta * theta + 1.0f));
                    tt = (theta < 0.0f) ? -tt : tt;
                    const float c = 1.0f / sqrtf(tt * tt + 1.0f), s = tt * c;
                    a00 -= tt * apq; a11 += tt * apq; a01 = 0.0f;
                    const float r0 = a02, r1 = a12;
                    a02 = c * r0 - s * r1; a12 = s * r0 + c * r1;
                    float t0, t1;
                    t0 = v00; t1 = v01; v00 = c * t0 - s * t1; v01 = s * t0 + c * t1;
                    t0 = v10; t1 = v11; v10 = c * t0 - s * t1; v11 = s * t0 + c * t1;
                    t0 = v20; t1 = v21; v20 = c * t0 - s * t1; v21 = s * t0 + c * t1;
                }
            }
            // (p,q) = (0,2), r = 1
            {
                const float apq = a02;
                if (fabsf(apq) > 1e-30f) {
                    const float theta = (a22 - a00) / (2.0f * apq);
                    float tt = 1.0f / (fabsf(theta) + sqrtf(theta * theta + 1.0f));
                    tt = (theta < 0.0f) ? -tt : tt;
                    const float c = 1.0f / sqrtf(tt * tt + 1.0f), s = tt * c;
                    a00 -= tt * apq; a22 += tt * apq; a02 = 0.0f;
                    const float r0 = a01, r1 = a12;
                    a01 = c * r0 - s * r1; a12 = s * r0 + c * r1;
                    float t0, t1;
                    t0 = v00; t1 = v02; v00 = c * t0 - s * t1; v02 = s * t0 + c * t1;
                    t0 = v10; t1 = v12; v10 = c * t0 - s * t1; v12 = s * t0 + c * t1;
                    t0 = v20; t1 = v22; v20 = c * t0 - s * t1; v22 = s * t0 + c * t1;
                }
            }
            // (p,q) = (1,2), r = 0
            {
                const float apq = a12;
                if (fabsf(apq) > 1e-30f) {
                    const float theta = (a22 - a11) / (2.0f * apq);
                    float tt = 1.0f / (fabsf(theta) + sqrtf(theta * theta + 1.0f));
                    tt = (theta < 0.0f) ? -tt : tt;
                    const float c = 1.0f / sqrtf(tt * tt + 1.0f), s = tt * c;
                    a11 -= tt * apq; a22 += tt * apq; a12 = 0.0f;
                    const float r0 = a01, r1 = a02;
                    a01 = c * r0 - s * r1; a02 = s * r0 + c * r1;
                    float t0, t1;
                    t0 = v01; t1 = v02; v01 = c * t0 - s * t1; v02 = s * t0 + c * t1;
                    t0 = v11; t1 = v12; v11 = c * t0 - s * t1; v12 = s * t0 + c * t1;
                    t0 = v21; t1 = v22; v21 = c * t0 - s * t1; v22 = s * t0 + c * t1;
                }
            }
        }

        // sort eigenpairs ascending (columns of V are eigenvectors)
        float e0 = a00, e1 = a11, e2 = a22;
        float c0x = v00, c0y = v10, c0z = v20;
        float c1x = v01, c1y = v11, c1z = v21;
        float c2x = v02, c2y = v12, c2z = v22;
        float te, tx, ty, tz;
        if (e0 > e1) { te = e0; e0 = e1; e1 = te; tx = c0x; ty = c0y; tz = c0z; c0x = c1x; c0y = c1y; c0z = c1z; c1x = tx; c1y = ty; c1z = tz; }
        if (e1 > e2) { te = e1; e1 = e2; e2 = te; tx = c1x; ty = c1y; tz = c1z; c1x = c2x; c1y = c2y; c1z = c2z; c2x = tx; c2y = ty; c2z = tz; }
        if (e0 > e1) { te = e0; e0 = e1; e1 = te; tx = c0x; ty = c0y; tz = c0z; c0x = c1x; c0y = c1y; c0z = c1z; c1x = tx; c1y = ty; c1z = tz; }

        float zx = c0x, zy = c0y, zz2 = c0z;   // smallest-eigenvalue direction
        float xx = c2x, xy = c2y, xz = c2z;    // largest-eigenvalue direction

        // majority-vote sign fixing
        int negx = 0, negz = 0;
        for (int j = 0; j < KNB; ++j) {
            const float4 nb = s_nb[m][j];
            negx += (nb.x * xx + nb.y * xy + nb.z * xz) < 0.0f;
            negz += (nb.x * zx + nb.y * zy + nb.z * zz2) < 0.0f;
        }
        if (negx > KNB - negx) { xx = -xx; xy = -xy; xz = -xz; }
        if (negz > KNB - negz) { zx = -zx; zy = -zy; zz2 = -zz2; }

        // y = cross(z, x)
        const float yx = zy * xz - zz2 * xy;
        const float yy = zz2 * xx - zx * xz;
        const float yz = zx * xy - zy * xx;

        // project to tangent plane, rescale to geodesic (Euclidean) length
        float* op = out + (((size_t)b * NPTS + row) * KNB) * 2;
        for (int j = 0; j < KNB; ++j) {
            const float4 nb = s_nb[m][j];
            const float lx = xx * nb.x + xy * nb.y + xz * nb.z;
            const float ly = yx * nb.x + yy * nb.y + yz * nb.z;
            const float pn2 = lx * lx + ly * ly;
            const float scale = (pn2 > 0.0f) ? nb.w / sqrtf(pn2) : 0.0f;
            op[2 * j + 0] = lx * scale;
            op[2 * j + 1] = ly * scale;
        }
    }
}

extern "C" void kernel_launch(void* const* d_in, const int* in_sizes, int n_in,
                              void* d_out, int out_size, void* d_ws, size_t ws_size,
                              hipStream_t stream) {
    (void)in_sizes; (void)n_in; (void)d_ws; (void)ws_size; (void)out_size;
    const float* coords = (const float*)d_in[0];   // (4, 4096, 3) f32
    // d_in[1] = n_neighbors (compile-time 32 here)
    float* out = (float*)d_out;                    // (4, 4096, 32, 2) f32

    dim3 grid(BATCH * (NPTS / 16));
    dim3 block(32);
    TangentProjections_89369679495272_kernel<<<grid, block, 0, stream>>>(coords, out);
}